// PointNetPPFeatureAggregation_59115929862336
// MI455X (gfx1250) — compile-verified
//
#include <hip/hip_runtime.h>

typedef float v2f __attribute__((ext_vector_type(2)));
typedef float v8f __attribute__((ext_vector_type(8)));

namespace {
constexpr int kB = 8;      // batches
constexpr int kN = 2048;   // points
constexpr int kF = 512;    // features
constexpr int kK = 32;     // neighbors
constexpr int kT = 16;     // WMMA tile M/N
constexpr int kWaves = 4;              // waves per block
constexpr int kCand  = kN / kWaves;    // candidates scanned per wave (512)
}

// One block (4 waves) per (batch, 16-query tile).
//  Stage 0: cooperatively stage the batch's 2048x3 points into LDS (24 KB).
//  Stage 1: each wave scans a disjoint 512-candidate range:
//           scores S(i,j) = -2*pi.pj + |pj|^2 via V_WMMA_F32_16X16X4_F32
//           (row-constant |pi|^2 omitted; per-row ordering unchanged),
//           per-query partial top-32 (replace-worst + rescan) in LDS.
//  Stage 2: merge the 4 partial lists (disjoint ranges -> no duplicates).
//  Stage 3: gather + max-pool features, float4 loads, L2-resident.
__launch_bounds__(kWaves * 32)
__global__ void pointnetpp_knn_maxpool(const float* __restrict__ x,
                                       const float* __restrict__ points,
                                       float* __restrict__ out) {
  __shared__ float s_pts[kN * 3];               // 24 KB staged points
  __shared__ float s_scores[kWaves][kT * kT];   //  4 KB score tiles
  __shared__ float s_topd[kWaves][kT][kK];      //  8 KB partial top-K dists
  __shared__ int   s_topi[kWaves][kT][kK];      //  8 KB partial top-K indices

  const int tid  = (int)threadIdx.x;
  const int wave = tid >> 5;
  const int lane = tid & 31;
  const int b    = (int)blockIdx.x / (kN / kT);
  const int q0   = ((int)blockIdx.x % (kN / kT)) * kT;

  // ---- Stage 0: points -> LDS (one pass of b128 copies) --------------------
  {
    const float4* src = (const float4*)(points + (size_t)b * kN * 3);
    float4*       dst = (float4*)s_pts;
    for (int i = tid; i < (kN * 3) / 4; i += kWaves * 32) dst[i] = src[i];
  }

  const int half = lane >> 4;   // which K-pair this lane holds
  const int sub  = lane & 15;   // A: row m / B: col n

  // init this wave's partial top-K lists
  if (lane < kT) {
#pragma unroll
    for (int k = 0; k < kK; ++k) {
      s_topd[wave][lane][k] = __builtin_huge_valf();
      s_topi[wave][lane][k] = 0;
    }
  }
  __syncthreads();

  // ---- A fragment: 16 queries x K=4 = (x, y, z, 1) -------------------------
  // f32 16x4 layout: lanes 0-15 hold K={0,1}, lanes 16-31 hold K={2,3}.
  v2f afrag;
  {
    const float qx = s_pts[(q0 + sub) * 3 + 0];
    const float qy = s_pts[(q0 + sub) * 3 + 1];
    const float qz = s_pts[(q0 + sub) * 3 + 2];
    afrag.x = half ? qz   : qx;
    afrag.y = half ? 1.0f : qy;
  }

  float wv = __builtin_huge_valf();  // current worst (max) in my list
  int   wi = 0;                      // its slot

  // ---- Stage 1: WMMA score tiles + partial selection -----------------------
  const int cbeg = wave * kCand;
  const int cend = cbeg + kCand;
  for (int c0 = cbeg; c0 < cend; c0 += kT) {
    // B fragment: col j = c0+n, K=4 rows = (-2x, -2y, -2z, |p|^2), from LDS
    const int   j   = c0 + sub;
    const float px  = s_pts[j * 3 + 0];
    const float py  = s_pts[j * 3 + 1];
    const float pz  = s_pts[j * 3 + 2];
    const float nrm = px * px + py * py + pz * pz;
    v2f bfrag;
    bfrag.x = half ? (-2.0f * pz) : (-2.0f * px);
    bfrag.y = half ? nrm          : (-2.0f * py);

    v8f cacc = {0.f, 0.f, 0.f, 0.f, 0.f, 0.f, 0.f, 0.f};
    cacc = __builtin_amdgcn_wmma_f32_16x16x4_f32(
        /*neg_a=*/false, afrag, /*neg_b=*/false, bfrag,
        /*c_mod=*/(short)0, cacc, /*reuse_a=*/false, /*reuse_b=*/false);

    // C/D layout: VGPR v -> M = v + 8*half, N = sub
#pragma unroll
    for (int v = 0; v < 8; ++v) {
      s_scores[wave][(v + (half << 3)) * kT + sub] = cacc[v];
    }
    // wave-local cross-lane LDS handoff: wave32 lockstep + DS completion wait
    asm volatile("s_wait_dscnt 0x0" ::: "memory");

    // 16 lanes each own one query row; merge 16 new candidates
    if (lane < kT) {
#pragma unroll 4
      for (int t = 0; t < kT; ++t) {
        const float d = s_scores[wave][lane * kT + t];
        if (d < wv) {
          s_topd[wave][lane][wi] = d;
          s_topi[wave][lane][wi] = c0 + t;
          float nwv = s_topd[wave][lane][0];
          int   nwi = 0;
          for (int k = 1; k < kK; ++k) {
            const float dk = s_topd[wave][lane][k];
            if (dk > nwv) { nwv = dk; nwi = k; }
          }
          wv = nwv;
          wi = nwi;
        }
      }
    }
    // DS ops from one wave stay in order; next iteration's stores cannot
    // bypass this iteration's scan loads.
  }
  __syncthreads();

  // ---- Stage 2: merge partial lists into wave 0's list ---------------------
  if (wave == 0 && lane < kT) {
    for (int w = 1; w < kWaves; ++w) {
#pragma unroll 4
      for (int k = 0; k < kK; ++k) {
        const float d = s_topd[w][lane][k];
        if (d < wv) {
          s_topd[0][lane][wi] = d;
          s_topi[0][lane][wi] = s_topi[w][lane][k];
          float nwv = s_topd[0][lane][0];
          int   nwi = 0;
          for (int k2 = 1; k2 < kK; ++k2) {
            const float dk = s_topd[0][lane][k2];
            if (dk > nwv) { nwv = dk; nwi = k2; }
          }
          wv = nwv;
          wi = nwi;
        }
      }
    }
  }
  __syncthreads();

  // ---- Stage 3: gather + max-pool (features are L2-resident) ---------------
  const float* xb = x   + (size_t)b * kN * kF;
  float*       ob = out + (size_t)b * kN * kF;
  const float NEG_INF = -__builtin_huge_valf();

  const int mq0 = wave * (kT / kWaves);          // 4 queries per wave
  for (int mq = mq0; mq < mq0 + (kT / kWaves); ++mq) {
    for (int ch = 0; ch < kF / (32 * 4); ++ch) { // 4 chunks of 128 floats
      const int f = ch * 128 + lane * 4;
      float4 acc = make_float4(NEG_INF, NEG_INF, NEG_INF, NEG_INF);
#pragma unroll 4
      for (int k = 0; k < kK; ++k) {
        const int jj = s_topi[0][mq][k];
        const float4 vx = *(const float4*)(xb + (size_t)jj * kF + f);
        acc.x = fmaxf(acc.x, vx.x);
        acc.y = fmaxf(acc.y, vx.y);
        acc.z = fmaxf(acc.z, vx.z);
        acc.w = fmaxf(acc.w, vx.w);
      }
      *(float4*)(ob + (size_t)(q0 + mq) * kF + f) = acc;
    }
  }
}

extern "C" void kernel_launch(void* const* d_in, const int* in_sizes, int n_in,
                              void* d_out, int out_size, void* d_ws, size_t ws_size,
                              hipStream_t stream) {
  (void)in_sizes; (void)n_in; (void)out_size; (void)d_ws; (void)ws_size;
  const float* x      = (const float*)d_in[0];   // [B, N, F]
  const float* points = (const float*)d_in[1];   // [B, N, 3]
  float*       out    = (float*)d_out;           // [B, N, F]

  const int blocks = kB * (kN / kT);             // 1024 blocks x 4 waves
  pointnetpp_knn_maxpool<<<blocks, kWaves * 32, 0, stream>>>(x, points, out);
}